// CubeSpherePadding2D_64304250356408
// MI455X (gfx1250) — compile-verified
//
#include <hip/hip_runtime.h>

// Cube-sphere halo padding, p=1.
// Input : [16, 64, 6, 96, 96] f32
// Output: [16, 64, 6, 98, 98] f32
//
// Pure data movement (~0.47 GB total traffic => ~20us at 23.3 TB/s HBM).
// No matrix math exists in this op, so WMMA does not apply. Strategy:
//   Phase 1: one full-bandwidth streaming pass writing the entire output as
//            float2 (b64 stores; rows are 8B-aligned since 98*4=392=49*8),
//            interior <- non-temporal input loads, ring <- 0.
//   Phase 2: 6 tiny per-face halo kernels launched sequentially so reads of
//            the output buffer observe exactly the state the reference's
//            sequential .at[].set() chain observes (corner propagation).

#define NB    16
#define NC    64
#define NFACE 6
#define H     96
#define W     96
#define HP    98
#define WP    98

constexpr unsigned PAIRS      = NB * NC;                      // 1024
constexpr unsigned FACE_ELEMS = HP * WP;                      // 9604
constexpr unsigned IN_FACE    = H * W;                        // 9216
constexpr unsigned OUT_TOTAL  = PAIRS * NFACE * FACE_ELEMS;   // 59,006,976
constexpr unsigned OUT_PAIRS2 = OUT_TOTAL / 2;                // 29,503,488 float2
constexpr unsigned ROW_F2     = WP / 2;                       // 49 float2 per row
constexpr unsigned HALO_TPP   = 4 * WP;                       // 392 threads/pair
constexpr unsigned HALO_TOTAL = PAIRS * HALO_TPP;             // 401,408

// ---------------------------------------------------------------------------
// Phase 1: interior copy + zero halo ring, vectorized float2 stores.
// ---------------------------------------------------------------------------
__global__ void __launch_bounds__(256)
fill_interior(const float* __restrict__ in, float2* __restrict__ out2) {
    unsigned k = blockIdx.x * blockDim.x + threadIdx.x;   // float2 index
    if (k >= OUT_PAIRS2) return;

    unsigned p  = k % ROW_F2;        // float2 position in row; x0 = 2p, x1 = 2p+1
    unsigned rr = k / ROW_F2;        // global row index = r*98 + y
    unsigned y  = rr % HP;
    unsigned r  = rr / HP;           // r = pair*6 + face

    float2 v = make_float2(0.0f, 0.0f);
    if (y - 1u < (unsigned)H) {      // interior row
        const float* src = in + r * IN_FACE + (y - 1u) * W;
        unsigned x0 = 2u * p;
        if (p == 0u) {
            // x0=0 halo, x1=1 -> input col 0
            v.y = __builtin_nontemporal_load(src);
        } else if (p == ROW_F2 - 1u) {
            // x0=96 -> input col 95, x1=97 halo
            v.x = __builtin_nontemporal_load(src + (W - 1u));
        } else {
            v.x = __builtin_nontemporal_load(src + (x0 - 1u));
            v.y = __builtin_nontemporal_load(src + x0);
        }
    }
    out2[k] = v;
}

// ---------------------------------------------------------------------------
// Phase 2: per-face halo stitching. Faces processed sequentially (one kernel
// per face, in reference order):
//   - earlier faces: fully halo-updated when read
//   - later faces:   phase-1 state (interior + zero ring) when read
// Within a face the 4 assignments read only OTHER faces, so they may run
// concurrently; column writes skip j in {0,97} so the row writes own the
// corners (their final reference value).
//
// Encoding per [face][a], a = 0:col x=97, 1:col x=0, 2:row y=0, 3:row y=97.
// Fields {srcFace, syMode, syK, sxMode, sxK}; mode 0 -> K, 1 -> j, 2 -> 97-j.
// ---------------------------------------------------------------------------
__device__ constexpr int kTbl[6][4][5] = {
    // face 0: O0[y][97]=O1[y][1]    O0[y][0]=O3[y][96]    O0[0][x]=O4[96][x]    O0[97][x]=O5[1][x]
    { {1,1,0, 0,1},  {3,1,0, 0,96}, {4,0,96, 1,0},  {5,0,1,  1,0} },
    // face 1: O1[y][97]=O2[y][1]    O1[y][0]=O0[y][96]    O1[0][x]=O4[97-x][96] O1[97][x]=O5[x][96]
    { {2,1,0, 0,1},  {0,1,0, 0,96}, {4,2,0,  0,96}, {5,1,0,  0,96} },
    // face 2: O2[y][97]=O3[y][1]    O2[y][0]=O1[y][96]    O2[0][x]=O4[1][97-x]  O2[97][x]=O5[96][97-x]
    { {3,1,0, 0,1},  {1,1,0, 0,96}, {4,0,1,  2,0},  {5,0,96, 2,0} },
    // face 3: O3[y][97]=O0[y][1]    O3[y][0]=O2[y][96]    O3[0][x]=O4[x][1]     O3[97][x]=O5[97-x][1]
    { {0,1,0, 0,1},  {2,1,0, 0,96}, {4,1,0,  0,1},  {5,2,0,  0,1} },
    // face 4: O4[y][97]=O1[1][97-y] O4[y][0]=O3[1][y]     O4[0][x]=O2[1][97-x]  O4[97][x]=O0[1][x]
    { {1,0,1, 2,0},  {3,0,1, 1,0},  {2,0,1,  2,0},  {0,0,1,  1,0} },
    // face 5: O5[y][97]=O1[96][y]   O5[y][0]=O3[96][97-y] O5[0][x]=O0[96][x]    O5[97][x]=O2[96][97-x]
    { {1,0,96,1,0},  {3,0,96,2,0},  {0,0,96, 1,0},  {2,0,96, 2,0} },
};

__device__ __forceinline__ unsigned coordOf(int mode, int K, unsigned j) {
    return (mode == 0) ? (unsigned)K : ((mode == 1) ? j : 97u - j);
}

template <int F>
__global__ void __launch_bounds__(256)
halo_face(float* __restrict__ out) {
    unsigned tid = blockIdx.x * blockDim.x + threadIdx.x;
    if (tid >= HALO_TOTAL) return;
    unsigned pair = tid / HALO_TPP;
    unsigned i    = tid - pair * HALO_TPP;
    unsigned a    = i / WP;          // which of the 4 halo assignments
    unsigned j    = i - a * WP;      // free index 0..97

    // columns span y=1..96 only; corners belong to the row assignments
    if (a < 2u && (j == 0u || j == 97u)) return;

    unsigned sf = (unsigned)kTbl[F][a][0];
    unsigned sy = coordOf(kTbl[F][a][1], kTbl[F][a][2], j);
    unsigned sx = coordOf(kTbl[F][a][3], kTbl[F][a][4], j);

    unsigned dy = (a < 2u) ? j : ((a == 2u) ? 0u : 97u);
    unsigned dx = (a == 0u) ? 97u : ((a == 1u) ? 0u : j);

    unsigned base = pair * (NFACE * FACE_ELEMS);
    out[base + (unsigned)F * FACE_ELEMS + dy * WP + dx] =
        out[base + sf * FACE_ELEMS + sy * WP + sx];
}

// ---------------------------------------------------------------------------
extern "C" void kernel_launch(void* const* d_in, const int* in_sizes, int n_in,
                              void* d_out, int out_size, void* d_ws, size_t ws_size,
                              hipStream_t stream) {
    (void)in_sizes; (void)n_in; (void)out_size; (void)d_ws; (void)ws_size;

    const float* in = (const float*)d_in[0];
    float* out = (float*)d_out;

    // Phase 1: full output written every call (interior + zero ring).
    {
        unsigned threads = 256;
        unsigned blocks  = (OUT_PAIRS2 + threads - 1) / threads;  // 115,248
        fill_interior<<<blocks, threads, 0, stream>>>(in, (float2*)out);
    }

    // Phase 2: sequential per-face halo stitching (order matters for corners).
    {
        unsigned threads = 256;
        unsigned blocks  = (HALO_TOTAL + threads - 1) / threads;  // 1,568
        halo_face<0><<<blocks, threads, 0, stream>>>(out);
        halo_face<1><<<blocks, threads, 0, stream>>>(out);
        halo_face<2><<<blocks, threads, 0, stream>>>(out);
        halo_face<3><<<blocks, threads, 0, stream>>>(out);
        halo_face<4><<<blocks, threads, 0, stream>>>(out);
        halo_face<5><<<blocks, threads, 0, stream>>>(out);
    }
}